// ColorReducer_39865886442290
// MI455X (gfx1250) — compile-verified
//
#include <hip/hip_runtime.h>

typedef float v2f __attribute__((ext_vector_type(2)));
typedef float v8f __attribute__((ext_vector_type(8)));

#define HW_  (512 * 512)
#define TPI  4   // 16-pixel tiles per wave iteration

__device__ __forceinline__ unsigned umin32(unsigned a, unsigned b) {
    return a < b ? a : b;
}

// D[color][pixel] = ||c||^2 + 8 - 2 p.c  via V_WMMA_F32_16X16X4_F32
//   A (16 colors x K4)  : (-2cr, -2cg, -2cb, ||c||^2+8)
//   B (K4 x 16 pixels)  : ( r  ,  g  ,  b  ,    1     )
// f32 A 16x4 layout: lanes 0-15 -> v0=K0,v1=K1 ; lanes 16-31 -> v0=K2,v1=K3
// f32 B 4x16 layout mirrors the K split (lanes 0-15: K0/K1, lanes 16-31: K2/K3)
// D layout: VGPR r -> color M=r (lanes 0-15) / M=r+8 (lanes 16-31), pixel N=lane%16
__global__ __launch_bounds__(256) void ColorReducer_wmma_kernel(
    const float* __restrict__ x,
    const float* __restrict__ pal,
    float* __restrict__ out,
    int numTiles)
{
    const int lane = threadIdx.x & 31;
    const int hi   = lane >> 4;              // 0: lanes 0-15, 1: lanes 16-31
    const int p    = lane & 15;
    const unsigned hb8 = (unsigned)(hi << 3);

    // ---- Loop-invariant palette A-matrices: 4 groups of 16 colors ----
    v2f A[4];
    #pragma unroll
    for (int g = 0; g < 4; ++g) {
        int c = g * 16 + p;
        float cr = pal[3 * c + 0];
        float cg = pal[3 * c + 1];
        float cb = pal[3 * c + 2];
        // +8 bias keeps all distances positive (p.c <= 3) so uint-order == float-order
        float n2 = fmaf(cr, cr, fmaf(cg, cg, fmaf(cb, cb, 8.0f)));
        A[g].x = hi ? (-2.0f * cb) : (-2.0f * cr);   // K2 : K0
        A[g].y = hi ? n2           : (-2.0f * cg);   // K3 : K1
    }

    const int wavesPerBlock = blockDim.x >> 5;
    const int gwave = blockIdx.x * wavesPerBlock + (threadIdx.x >> 5);
    const int step  = gridDim.x * wavesPerBlock * TPI;

    for (int base = gwave * TPI; base < numTiles; base += step) {
        int P0  = base * 16;                  // flattened pixel index b*HW + hw
        int b   = P0 / HW_;
        int hw0 = P0 - b * HW_;               // 64 consecutive pixels, same image
        const float* xb = x   + (size_t)b * 3 * HW_ + hw0;
        float*       ob = out + (size_t)b * 3 * HW_ + hw0;

        v8f D[TPI][4];
        #pragma unroll
        for (int j = 0; j < TPI; ++j) {
            int o = j * 16 + p;
            float gch = xb[HW_ + o];                  // green plane (coalesced, dup across halves)
            float bv0 = xb[(hi ? 2 * HW_ : 0) + o];   // red (low lanes) / blue (high lanes)
            float bv1 = hi ? 1.0f : gch;              // green (low) / constant 1 (high)
            v2f B;
            B.x = bv0;
            B.y = bv1;
            #pragma unroll
            for (int g = 0; g < 4; ++g) {
                v8f cz = {};
                D[j][g] = __builtin_amdgcn_wmma_f32_16x16x4_f32(
                    false, A[g], false, B, (short)0, cz, false, false);
            }
        }

        #pragma unroll
        for (int j = 0; j < TPI; ++j) {
            // Per-lane argmin over this lane's 32 colors: pack index into low 6
            // mantissa bits (<= 2^-17 relative perturbation) and v_min_u32-reduce.
            unsigned best = 0xFFFFFFFFu;
            #pragma unroll
            for (int g = 0; g < 4; ++g) {
                #pragma unroll
                for (int r = 0; r < 8; ++r) {
                    unsigned K   = (unsigned)(g * 16 + r);
                    unsigned key = (__float_as_uint(D[j][g][r]) & ~63u) | K | hb8;
                    best = umin32(best, key);
                }
            }
            // Merge the two half-wave color sets: SWAPX16 (xor 0x10) ds_swizzle.
            unsigned other = (unsigned)__builtin_amdgcn_ds_swizzle((int)best, 0x401F);
            best = umin32(best, other);
            int idx = (int)(best & 63u);

            // Write the chosen palette color, planar layout.
            int o = j * 16 + p;
            float cv = pal[3 * idx + (hi ? 2 : 0)];
            ob[(hi ? 2 * HW_ : 0) + o] = cv;          // red (low) / blue (high)
            if (!hi) ob[HW_ + o] = pal[3 * idx + 1];  // green (low lanes only)
        }
    }
}

extern "C" void kernel_launch(void* const* d_in, const int* in_sizes, int n_in,
                              void* d_out, int out_size, void* d_ws, size_t ws_size,
                              hipStream_t stream) {
    const float* x   = (const float*)d_in[0];   // (8, 3, 512, 512) f32
    const float* pal = (const float*)d_in[1];   // (64, 3) f32
    float* out = (float*)d_out;                 // (8, 3, 512, 512) f32

    int numPix   = in_sizes[0] / 3;             // 2,097,152
    int numTiles = numPix / 16;                 // 131,072
    int wavesNeeded = (numTiles + TPI - 1) / TPI;       // 32,768 waves
    int blocks = (wavesNeeded + 7) / 8;                 // 256 threads = 8 waves/block
    ColorReducer_wmma_kernel<<<blocks, 256, 0, stream>>>(x, pal, out, numTiles);
}